// DNC_618475290988
// MI455X (gfx1250) — compile-verified
//
#include <hip/hip_runtime.h>
#include <math.h>

// Problem sizes (match reference)
#define T_STEPS 64
#define XDIM    512
#define NMEM    2048
#define WDIM    128
#define RHEADS  4
#define YDIM    512
#define GDIM    2048   // 4*X

typedef float v2f __attribute__((ext_vector_type(2)));
typedef float v8f __attribute__((ext_vector_type(8)));

#if __has_builtin(__builtin_amdgcn_wmma_f32_16x16x4_f32)
#define USE_WMMA_F32X4 1
#endif

// ---------------- workspace layout (floats) ----------------
constexpr size_t OFF_L      = 0;                                // 2048*2048 (zeroed)
constexpr size_t OFF_USAGE  = OFF_L      + (size_t)NMEM*NMEM;   // 2048 (zeroed)
constexpr size_t OFF_WW     = OFF_USAGE  + NMEM;                // 2048 (zeroed)
constexpr size_t OFF_PREC   = OFF_WW     + NMEM;                // 2048 (zeroed)
constexpr size_t OFF_RW     = OFF_PREC   + NMEM;                // 4*2048 (zeroed)
constexpr size_t OFF_H      = OFF_RW     + RHEADS*NMEM;         // 2*512 (zeroed)
constexpr size_t OFF_C      = OFF_H      + 2*XDIM;              // 2*512 (zeroed)
constexpr size_t ZERO_FLOATS= OFF_C      + 2*XDIM;              // everything above zeroed per call
constexpr size_t OFF_M      = ZERO_FLOATS;                      // 2048*128 (copied from M0)
constexpr size_t OFF_ZX     = OFF_M      + (size_t)NMEM*WDIM;   // 64*2048
constexpr size_t OFF_Z      = OFF_ZX     + (size_t)T_STEPS*GDIM;// 2048
constexpr size_t OFF_RBUF   = OFF_Z      + GDIM;                // 64*512
constexpr size_t OFF_READS  = OFF_RBUF   + (size_t)T_STEPS*XDIM;// 64*512
constexpr size_t OFF_RK     = OFF_READS  + (size_t)T_STEPS*XDIM;// 512 read keys
constexpr size_t OFF_WK     = OFF_RK     + RHEADS*WDIM;         // 128
constexpr size_t OFF_ERASE  = OFF_WK     + WDIM;                // 128
constexpr size_t OFF_WVEC   = OFF_ERASE  + WDIM;                // 128
constexpr size_t OFF_RS     = OFF_WVEC   + WDIM;                // 4
constexpr size_t OFF_FG     = OFF_RS     + RHEADS;              // 4
constexpr size_t OFF_SCAL   = OFF_FG     + RHEADS;              // 4: [wstrength, alloc_gate, write_gate, |wk|]
constexpr size_t OFF_RM     = OFF_SCAL   + 4;                   // 4 (3 used)
constexpr size_t OFF_RKN    = OFF_RM     + 4;                   // 4
constexpr size_t OFF_ALLOC  = OFF_RKN    + 4;                   // 2048
constexpr size_t OFF_ACOS   = OFF_ALLOC  + NMEM;                // 2048
constexpr size_t OFF_RNORM  = OFF_ACOS   + NMEM;                // 2048
constexpr size_t OFF_RDOTS  = OFF_RNORM  + NMEM;                // 2048*4
constexpr size_t OFF_RCW    = OFF_RDOTS  + (size_t)NMEM*RHEADS; // 4*2048
constexpr size_t OFF_BW     = OFF_RCW    + RHEADS*NMEM;         // 4*2048
constexpr size_t OFF_FWPART = OFF_BW     + RHEADS*NMEM;         // 128*4*2048 deterministic partials
constexpr size_t WS_FLOATS  = OFF_FWPART + (size_t)(NMEM/16)*RHEADS*NMEM;

// ---------------- small device helpers ----------------
__device__ __forceinline__ float sigf(float x)   { return 1.f/(1.f+expf(-x)); }
__device__ __forceinline__ float splusf(float x) { return fmaxf(x,0.f)+log1pf(expf(-fabsf(x))); }

template<int BS>
__device__ __forceinline__ float blkSum(float v, float* s){
  int t=threadIdx.x; s[t]=v; __syncthreads();
  for(int o=BS>>1;o>0;o>>=1){ if(t<o) s[t]+=s[t+o]; __syncthreads(); }
  float r=s[0]; __syncthreads(); return r;
}
template<int BS>
__device__ __forceinline__ float blkMax(float v, float* s){
  int t=threadIdx.x; s[t]=v; __syncthreads();
  for(int o=BS>>1;o>0;o>>=1){ if(t<o) s[t]=fmaxf(s[t],s[t+o]); __syncthreads(); }
  float r=s[0]; __syncthreads(); return r;
}

// ---- CDNA5 async global->LDS copy (ASYNCcnt path, CDNA5 ISA) ----
// Generic pointers to LDS carry the LDS offset in their low 32 bits (aperture form),
// which is exactly the VDST operand of GLOBAL_LOAD_ASYNC_TO_LDS (the VGPR giving the LDS destination address).
__device__ __forceinline__ void async_b128_to_lds(void* lds_dst, const void* gsrc){
  unsigned lds_off = (unsigned)(uintptr_t)lds_dst;
  asm volatile("global_load_async_to_lds_b128 %0, %1, off"
               :: "v"(lds_off), "v"(gsrc) : "memory");
}
__device__ __forceinline__ void wait_async_lds(){
  asm volatile("s_wait_asynccnt 0x0" ::: "memory");
}

// ---------------- WMMA f32 16x16x4 tile GEMM ----------------
// A: [M,K] row-major (lda), B: [K,N] row-major (ldb), acc = 16x16 f32 tile.
__device__ __forceinline__ void gemm_tile(const float* __restrict__ A, int lda,
                                          const float* __restrict__ B, int ldb,
                                          int row0, int col0, int K, v8f& acc)
{
  const int lane = threadIdx.x;
  const int half = lane >> 4;
  const int lr   = lane & 15;
#ifdef USE_WMMA_F32X4
  for (int k0 = 0; k0 < K; k0 += 4) {
    v2f a, b;
    a.x = A[(size_t)(row0+lr)*lda + k0 + 2*half + 0];
    a.y = A[(size_t)(row0+lr)*lda + k0 + 2*half + 1];
    b.x = B[(size_t)(k0 + 2*half + 0)*ldb + col0 + lr];
    b.y = B[(size_t)(k0 + 2*half + 1)*ldb + col0 + lr];
    acc = __builtin_amdgcn_wmma_f32_16x16x4_f32(false, a, false, b, (short)0, acc, false, false);
  }
#else
  const int col = col0 + lr, rbase = row0 + half*8;
  for (int k = 0; k < K; ++k) {
    float bv = B[(size_t)k*ldb + col];
    #pragma unroll
    for (int e = 0; e < 8; ++e) acc[e] += A[(size_t)(rbase+e)*lda + k] * bv;
  }
#endif
}

// Zx = inputs[64x512] @ Wx[512x2048] + b   (hoisted out of the recurrence)
__global__ __launch_bounds__(32) void k_zx_gemm(const float* __restrict__ inp,
                                                const float* __restrict__ Wx,
                                                const float* __restrict__ bias,
                                                float* __restrict__ Zx)
{
  const int col0 = blockIdx.x*16, row0 = blockIdx.y*16;
  v8f acc = {};
  gemm_tile(inp, XDIM, Wx, GDIM, row0, col0, XDIM, acc);
  const int lane = threadIdx.x;
  const int col = col0 + (lane & 15), rbase = row0 + ((lane >> 4) << 3);
  const float bv = bias[col];
  #pragma unroll
  for (int e = 0; e < 8; ++e) Zx[(size_t)(rbase+e)*GDIM + col] = acc[e] + bv;
}

// Y = Rbuf[64x512]@VM[512x512] + Reads[64x512]@RM[512x512]   (deferred: y has no feedback)
__global__ __launch_bounds__(32) void k_out_gemm(const float* __restrict__ Rbuf,
                                                 const float* __restrict__ Reads,
                                                 const float* __restrict__ VM,
                                                 const float* __restrict__ RM,
                                                 float* __restrict__ out)
{
  const int col0 = blockIdx.x*16, row0 = blockIdx.y*16;
  v8f acc = {};
  gemm_tile(Rbuf,  XDIM,         VM, YDIM, row0, col0, XDIM,         acc);
  gemm_tile(Reads, RHEADS*WDIM,  RM, YDIM, row0, col0, RHEADS*WDIM,  acc);
  const int lane = threadIdx.x;
  const int col = col0 + (lane & 15), rbase = row0 + ((lane >> 4) << 3);
  #pragma unroll
  for (int e = 0; e < 8; ++e) out[(size_t)(rbase+e)*YDIM + col] = acc[e];
}

// ---------------- per-step kernels ----------------
// z = Zx[t] + h @ Wh   (Wh 4MB/step, L2-resident); h staged via async-to-LDS
__global__ __launch_bounds__(128) void k_zh(int t, const float* __restrict__ Zx,
                                            const float* __restrict__ Wh,
                                            const float* __restrict__ h_in,
                                            float* __restrict__ z)
{
  __shared__ float s_h[XDIM];
  const int tid = threadIdx.x;
  async_b128_to_lds(&s_h[tid*4], h_in + tid*4);   // 128 lanes x 16B = 512 floats
  wait_async_lds();
  __syncthreads();
  const int col = blockIdx.x*128 + tid;
  float acc = Zx[(size_t)t*GDIM + col];
  const float* wp = Wh + col;
  for (int x = 0; x < XDIM; ++x) acc += s_h[x] * wp[(size_t)x*GDIM];
  z[col] = acc;
}

// LSTMBlockCell gates (i,j,f,o; forget_bias=1)
__global__ __launch_bounds__(512) void k_gates(int t, const float* __restrict__ z,
                                               const float* __restrict__ c_in,
                                               float* __restrict__ c_out,
                                               float* __restrict__ h_out,
                                               float* __restrict__ Rbuf)
{
  const int k = threadIdx.x;
  const float zi = z[k], zj = z[k+XDIM], zf = z[k+2*XDIM], zo = z[k+3*XDIM];
  const float cn = sigf(zf + 1.f)*c_in[k] + sigf(zi)*tanhf(zj);
  const float hn = sigf(zo)*tanhf(cn);
  c_out[k] = cn; h_out[k] = hn;
  Rbuf[(size_t)t*XDIM + k] = hn;     // r_t, consumed by the deferred output GEMM
}

// All interface projections in one kernel (910 outputs, segment-routed)
__global__ __launch_bounds__(128) void k_interface(const float* __restrict__ h,
    const float* IRK, const float* IRS, const float* IWK, const float* IWS,
    const float* IEM, const float* IWM, const float* IFG, const float* IAG,
    const float* IWG, const float* IRM,
    float* rk, float* rs, float* wk, float* erase, float* wvec,
    float* fg, float* scal, float* rm)
{
  __shared__ float s_h[XDIM];
  const int tid = threadIdx.x;
  async_b128_to_lds(&s_h[tid*4], h + tid*4);
  wait_async_lds();
  __syncthreads();
  const int g = blockIdx.x*128 + tid;
  const float* Wp = nullptr; int ldw = 0, col = 0, kind = 0; float* outp = nullptr;
  if      (g < 512){ Wp=IRK; ldw=512; col=g;      outp=rk+g;        }
  else if (g < 516){ Wp=IRS; ldw=4;   col=g-512;  outp=rs+col;  kind=1; }
  else if (g < 644){ Wp=IWK; ldw=128; col=g-516;  outp=wk+col;      }
  else if (g < 645){ Wp=IWS; ldw=1;   col=0;      outp=scal+0; kind=1; }
  else if (g < 773){ Wp=IEM; ldw=128; col=g-645;  outp=erase+col; kind=2; }
  else if (g < 901){ Wp=IWM; ldw=128; col=g-773;  outp=wvec+col;    }
  else if (g < 905){ Wp=IFG; ldw=4;   col=g-901;  outp=fg+col;  kind=2; }
  else if (g < 906){ Wp=IAG; ldw=1;   col=0;      outp=scal+1; kind=2; }
  else if (g < 907){ Wp=IWG; ldw=1;   col=0;      outp=scal+2; kind=2; }
  else if (g < 910){ Wp=IRM; ldw=3;   col=g-907;  outp=rm+col;      }
  if (Wp) {
    float acc = 0.f;
    for (int x = 0; x < XDIM; ++x) acc += s_h[x] * Wp[(size_t)x*ldw + col];
    if      (kind == 1) acc = 1.f + splusf(acc);   // strengths
    else if (kind == 2) acc = sigf(acc);           // gates / erase
    *outp = acc;
  }
}

// key norms + read_modes softmax
__global__ __launch_bounds__(128) void k_iface_post(const float* __restrict__ wk,
                                                    const float* __restrict__ rk,
                                                    float* scal, float* rm, float* rkn)
{
  __shared__ float sred[128];
  const int t = threadIdx.x;
  float s = blkSum<128>(wk[t]*wk[t], sred);
  if (t == 0) scal[3] = sqrtf(s);
  for (int r = 0; r < RHEADS; ++r) {
    const float v = rk[r*WDIM + t];
    float s2 = blkSum<128>(v*v, sred);
    if (t == 0) rkn[r] = sqrtf(s2);
  }
  if (t == 0) {
    const float a = rm[0], b = rm[1], c = rm[2];
    const float mx = fmaxf(a, fmaxf(b, c));
    const float ea = expf(a-mx), eb = expf(b-mx), ec = expf(c-mx);
    const float ss = ea + eb + ec;
    rm[0] = ea/ss; rm[1] = eb/ss; rm[2] = ec/ss;
  }
}

__device__ __forceinline__ void cswap(float& ka, int& va, float& kb, int& vb, unsigned ddd){
  const bool gt = (ka > kb) || (ka == kb && va > vb);   // lexicographic -> stable sort
  if (gt == (ddd != 0u)) {
    float tk = ka; ka = kb; kb = tk;
    int   tv = va; va = vb; vb = tv;
  }
}

// usage update + stable ascending bitonic argsort + exclusive cumprod -> allocation weights
__global__ __launch_bounds__(1024) void k_usage_alloc(const float* __restrict__ fg,
    const float* __restrict__ rw, float* __restrict__ usage,
    const float* __restrict__ ww, float* __restrict__ alloc)
{
  __shared__ float s_key[NMEM];
  __shared__ int   s_val[NMEM];
  __shared__ float s_scan[NMEM];
  const int tid = threadIdx.x;
  const float fg0 = fg[0], fg1 = fg[1], fg2 = fg[2], fg3 = fg[3];
  for (int e = tid; e < NMEM; e += 1024) {
    const float ret = (1.f - fg0*rw[e]) * (1.f - fg1*rw[NMEM+e])
                    * (1.f - fg2*rw[2*NMEM+e]) * (1.f - fg3*rw[3*NMEM+e]);
    const float u = usage[e], w = ww[e];
    const float un = ret * (u + w - u*w);
    usage[e] = un; s_key[e] = un; s_val[e] = e;
  }
  __syncthreads();
  for (unsigned size = 2; size < NMEM; size <<= 1) {
    const unsigned ddd = 1u ^ ((((unsigned)tid & (size >> 1)) != 0u) ? 1u : 0u);
    for (unsigned stride = size >> 1; stride > 0; stride >>= 1) {
      __syncthreads();
      const unsigned pos = 2u*(unsigned)tid - ((unsigned)tid & (stride - 1u));
      cswap(s_key[pos], s_val[pos], s_key[pos+stride], s_val[pos+stride], ddd);
    }
  }
  for (unsigned stride = NMEM/2; stride > 0; stride >>= 1) {
    __syncthreads();
    const unsigned pos = 2u*(unsigned)tid - ((unsigned)tid & (stride - 1u));
    cswap(s_key[pos], s_val[pos], s_key[pos+stride], s_val[pos+stride], 1u);
  }
  __syncthreads();
  s_scan[tid] = s_key[tid]; s_scan[tid+1024] = s_key[tid+1024];
  __syncthreads();
  for (int off = 1; off < NMEM; off <<= 1) {     // inclusive product scan
    const int e1 = tid + 1024;
    const float a0 = (tid >= off) ? s_scan[tid-off]*s_scan[tid] : s_scan[tid];
    const float a1 = (e1  >= off) ? s_scan[e1-off]*s_scan[e1]   : s_scan[e1];
    __syncthreads();
    s_scan[tid] = a0; s_scan[e1] = a1;
    __syncthreads();
  }
  for (int e = tid; e < NMEM; e += 1024) {
    const float cp = (e == 0) ? 1.f : s_scan[e-1];
    alloc[s_val[e]] = (1.f - s_key[e]) * cp;
  }
}

// pre-write cosine with the write key:  acos[n] = beta_w * (M[n].wk)/(|M[n]||wk|)
__global__ __launch_bounds__(128) void k_wcw_dot(const float* __restrict__ M,
                                                 const float* __restrict__ wk,
                                                 const float* __restrict__ scal,
                                                 float* __restrict__ acos_)
{
  __shared__ float sred[128];
  const int n = blockIdx.x, w = threadIdx.x;
  const float m = M[(size_t)n*WDIM + w];
  const float dot = blkSum<128>(m*wk[w], sred);
  const float n2  = blkSum<128>(m*m,    sred);
  if (w == 0) acos_[n] = scal[0] * dot / (sqrtf(n2) * scal[3]);
}

// write softmax -> ww -> precedence update (single workgroup)
__global__ __launch_bounds__(1024) void k_wsoftmax(const float* __restrict__ acos_,
                                                   const float* __restrict__ alloc,
                                                   const float* __restrict__ scal,
                                                   float* __restrict__ ww,
                                                   float* __restrict__ prec)
{
  __shared__ float sred[1024];
  const int t = threadIdx.x;
  const float a0 = acos_[t], a1 = acos_[t+1024];
  const float mx = blkMax<1024>(fmaxf(a0, a1), sred);
  const float e0 = expf(a0-mx), e1 = expf(a1-mx);
  const float s  = blkSum<1024>(e0+e1, sred);
  const float ga = scal[1], gw = scal[2];
  const float w0 = (ga*alloc[t]      + (1.f-ga)*(e0/s)) * gw;
  const float w1 = (ga*alloc[t+1024] + (1.f-ga)*(e1/s)) * gw;
  ww[t] = w0; ww[t+1024] = w1;
  const float sw = blkSum<1024>(w0+w1, sred);
  prec[t]      = prec[t]     *(1.f-sw) + w0;
  prec[t+1024] = prec[t+1024]*(1.f-sw) + w1;
}

// M write (erase+add), fused with row norms + dots against the 4 read keys (single M pass)
__global__ __launch_bounds__(128) void k_write_M(float* __restrict__ M,
    const float* __restrict__ ww, const float* __restrict__ erase,
    const float* __restrict__ wvec, const float* __restrict__ rk,
    float* __restrict__ rnorm, float* __restrict__ rdots)
{
  __shared__ float sred[128];
  const int n = blockIdx.x, w = threadIdx.x;
  const float wwn = ww[n];
  const size_t idx = (size_t)n*WDIM + w;
  const float m  = M[idx];
  const float mn = m*(1.f - wwn*erase[w]) + wwn*wvec[w];
  M[idx] = mn;
  const float n2 = blkSum<128>(mn*mn, sred);
  const float d0 = blkSum<128>(mn*rk[0*WDIM+w], sred);
  const float d1 = blkSum<128>(mn*rk[1*WDIM+w], sred);
  const float d2 = blkSum<128>(mn*rk[2*WDIM+w], sred);
  const float d3 = blkSum<128>(mn*rk[3*WDIM+w], sred);
  if (w == 0) {
    rnorm[n] = sqrtf(n2);
    rdots[n*4+0] = d0; rdots[n*4+1] = d1; rdots[n*4+2] = d2; rdots[n*4+3] = d3;
  }
}

// read content softmax over N per head
__global__ __launch_bounds__(1024) void k_rsoftmax(const float* __restrict__ rdots,
                                                   const float* __restrict__ rnorm,
                                                   const float* __restrict__ rs,
                                                   const float* __restrict__ rkn,
                                                   float* __restrict__ rcw)
{
  __shared__ float sred[1024];
  const int t = threadIdx.x;
  for (int r = 0; r < RHEADS; ++r) {
    const float a0 = rs[r]*rdots[t*4+r]        / (rnorm[t]*rkn[r]);
    const float a1 = rs[r]*rdots[(t+1024)*4+r] / (rnorm[t+1024]*rkn[r]);
    const float mx = blkMax<1024>(fmaxf(a0, a1), sred);
    const float e0 = expf(a0-mx), e1 = expf(a1-mx);
    const float s  = blkSum<1024>(e0+e1, sred);
    rcw[r*NMEM+t] = e0/s; rcw[r*NMEM+t+1024] = e1/s;
  }
}

// The 16MB hot loop: fused L update + fw/bw matvecs, one read + one write of L per step.
// Block owns a 16-row strip: bw exact in-block; fw written as deterministic per-block partials.
__global__ __launch_bounds__(256) void k_L_fwbw(float* __restrict__ L,
    const float* __restrict__ ww, const float* __restrict__ prec,
    const float* __restrict__ rw, float* __restrict__ fwpart, float* __restrict__ bw)
{
  __shared__ float s_wwi[16], s_rwA[RHEADS][16];
  __shared__ float s_wwj[16], s_precj[16], s_rwB[RHEADS][16];
  __shared__ float tile[16][17];
  __shared__ float s_bwacc[RHEADS][16];
  const int tid = threadIdx.x;
  const int i0  = blockIdx.x*16;
  if (tid < 16) s_wwi[tid] = ww[i0+tid];
  if (tid < 64) { const int r = tid>>4, q = tid&15;
                  s_rwA[r][q] = rw[r*NMEM + i0 + q]; s_bwacc[r][q] = 0.f; }
  __syncthreads();
  const int di = tid>>4, dj = tid&15;
  for (int j0 = 0; j0 < NMEM; j0 += 16) {
    if (tid < 16) { s_wwj[tid] = ww[j0+tid]; s_precj[tid] = prec[j0+tid]; }
    else if (tid >= 32 && tid < 96) { const int u = tid-32, r = u>>4, q = u&15;
                                      s_rwB[r][q] = rw[r*NMEM + j0 + q]; }
    if (j0 + 16 < NMEM)                                  // hide L2 latency for next tile
      __builtin_prefetch(&L[(size_t)(i0+di)*NMEM + j0 + 16 + dj], 0, 0);
    __syncthreads();
    const size_t lidx = (size_t)(i0+di)*NMEM + j0 + dj;
    const float l  = L[lidx];
    const float ln = l*(1.f - s_wwi[di] - s_wwj[dj]) + s_wwi[di]*s_precj[dj];
    L[lidx] = ln;
    tile[di][dj] = ln;
    __syncthreads();
    if (tid < 64) {                       // fw partial for this row-strip, this j-tile
      const int r = tid>>4, j = tid&15;
      float a = 0.f;
      #pragma unroll
      for (int q = 0; q < 16; ++q) a += s_rwA[r][q]*tile[q][j];
      fwpart[((size_t)blockIdx.x*RHEADS + r)*NMEM + j0 + j] = a;
    } else if (tid < 128) {               // bw accumulates across all j-tiles
      const int u = tid-64, r = u>>4, ii = u&15;
      float a = 0.f;
      #pragma unroll
      for (int q = 0; q < 16; ++q) a += s_rwB[r][q]*tile[ii][q];
      s_bwacc[r][ii] += a;
    }
    __syncthreads();
  }
  if (tid < 64) { const int r = tid>>4, q = tid&15; bw[r*NMEM + i0 + q] = s_bwacc[r][q]; }
}

// rw = rm0*bw + rm1*rcw + rm2*fw (fw reduced from deterministic partials here);
// reads[r] = rw[r] @ M
__global__ __launch_bounds__(128) void k_read_update(int t, const float* __restrict__ rm,
    const float* __restrict__ bw, const float* __restrict__ rcw,
    const float* __restrict__ fwpart, float* __restrict__ rw,
    const float* __restrict__ M, float* __restrict__ Reads)
{
  __shared__ float s_rw[NMEM];
  const int r = blockIdx.x, tid = threadIdx.x;
  const float m0 = rm[0], m1 = rm[1], m2 = rm[2];
  for (int q = 0; q < NMEM/128; ++q) {
    const int n = q*128 + tid;
    float f = 0.f;
    for (int b = 0; b < NMEM/16; ++b) f += fwpart[((size_t)b*RHEADS + r)*NMEM + n];
    const float v = m0*bw[r*NMEM+n] + m1*rcw[r*NMEM+n] + m2*f;
    rw[r*NMEM+n] = v; s_rw[n] = v;
  }
  __syncthreads();
  float acc = 0.f;
  for (int n = 0; n < NMEM; ++n) acc += s_rw[n]*M[(size_t)n*WDIM + tid];
  Reads[(size_t)t*(RHEADS*WDIM) + r*WDIM + tid] = acc;
}

// ---------------- host launcher ----------------
extern "C" void kernel_launch(void* const* d_in, const int* in_sizes, int n_in,
                              void* d_out, int out_size, void* d_ws, size_t ws_size,
                              hipStream_t stream)
{
  (void)in_sizes; (void)n_in; (void)out_size; (void)ws_size;
  const float* inp = (const float*)d_in[0];
  const float* Wx  = (const float*)d_in[1];
  const float* Wh  = (const float*)d_in[2];
  const float* b   = (const float*)d_in[3];
  const float* IRK = (const float*)d_in[4];
  const float* IRS = (const float*)d_in[5];
  const float* IWK = (const float*)d_in[6];
  const float* IWS = (const float*)d_in[7];
  const float* IEM = (const float*)d_in[8];
  const float* IWM = (const float*)d_in[9];
  const float* IFG = (const float*)d_in[10];
  const float* IAG = (const float*)d_in[11];
  const float* IWG = (const float*)d_in[12];
  const float* IRM = (const float*)d_in[13];
  const float* VM  = (const float*)d_in[14];
  const float* RM  = (const float*)d_in[15];
  const float* M0  = (const float*)d_in[16];

  float* ws     = (float*)d_ws;
  float* Lb     = ws + OFF_L;
  float* usage  = ws + OFF_USAGE;
  float* wwb    = ws + OFF_WW;
  float* precb  = ws + OFF_PREC;
  float* rwb    = ws + OFF_RW;
  float* Hb     = ws + OFF_H;
  float* Cb     = ws + OFF_C;
  float* Mb     = ws + OFF_M;
  float* Zx     = ws + OFF_ZX;
  float* zb     = ws + OFF_Z;
  float* Rbuf   = ws + OFF_RBUF;
  float* Reads  = ws + OFF_READS;
  float* rk     = ws + OFF_RK;
  float* wk     = ws + OFF_WK;
  float* erasev = ws + OFF_ERASE;
  float* wvec   = ws + OFF_WVEC;
  float* rs     = ws + OFF_RS;
  float* fgv    = ws + OFF_FG;
  float* scal   = ws + OFF_SCAL;
  float* rmv    = ws + OFF_RM;
  float* rkn    = ws + OFF_RKN;
  float* allocv = ws + OFF_ALLOC;
  float* acosv  = ws + OFF_ACOS;
  float* rnorm  = ws + OFF_RNORM;
  float* rdots  = ws + OFF_RDOTS;
  float* rcwb   = ws + OFF_RCW;
  float* bwb    = ws + OFF_BW;
  float* fwpart = ws + OFF_FWPART;

  // per-call deterministic state init (graph-capturable nodes)
  hipMemsetAsync(ws, 0, ZERO_FLOATS*sizeof(float), stream);
  hipMemcpyAsync(Mb, M0, (size_t)NMEM*WDIM*sizeof(float), hipMemcpyDeviceToDevice, stream);

  // hoisted input-projection GEMM (WMMA)
  k_zx_gemm<<<dim3(GDIM/16, T_STEPS/16), 32, 0, stream>>>(inp, Wx, b, Zx);

  for (int t = 0; t < T_STEPS; ++t) {
    float* h_in  = Hb + (t&1)*XDIM;  float* h_out = Hb + ((t+1)&1)*XDIM;
    float* c_in  = Cb + (t&1)*XDIM;  float* c_out = Cb + ((t+1)&1)*XDIM;
    k_zh        <<<16, 128, 0, stream>>>(t, Zx, Wh, h_in, zb);
    k_gates     <<<1, 512, 0, stream>>>(t, zb, c_in, c_out, h_out, Rbuf);
    k_interface <<<8, 128, 0, stream>>>(h_out, IRK, IRS, IWK, IWS, IEM, IWM, IFG, IAG,
                                        IWG, IRM, rk, rs, wk, erasev, wvec, fgv, scal, rmv);
    k_iface_post<<<1, 128, 0, stream>>>(wk, rk, scal, rmv, rkn);
    k_usage_alloc<<<1, 1024, 0, stream>>>(fgv, rwb, usage, wwb, allocv);
    k_wcw_dot   <<<NMEM, 128, 0, stream>>>(Mb, wk, scal, acosv);
    k_wsoftmax  <<<1, 1024, 0, stream>>>(acosv, allocv, scal, wwb, precb);
    k_write_M   <<<NMEM, 128, 0, stream>>>(Mb, wwb, erasev, wvec, rk, rnorm, rdots);
    k_rsoftmax  <<<1, 1024, 0, stream>>>(rdots, rnorm, rs, rkn, rcwb);
    k_L_fwbw    <<<NMEM/16, 256, 0, stream>>>(Lb, wwb, precb, rwb, fwpart, bwb);
    k_read_update<<<RHEADS, 128, 0, stream>>>(t, rmv, bwb, rcwb, fwpart, rwb, Mb, Reads);
  }

  // deferred output GEMM (WMMA)
  k_out_gemm<<<dim3(YDIM/16, T_STEPS/16), 32, 0, stream>>>(Rbuf, Reads, VM, RM, (float*)d_out);
}